// DMM_79800492360052
// MI455X (gfx1250) — compile-verified
//
#include <hip/hip_runtime.h>
#include <math.h>

typedef __attribute__((ext_vector_type(2))) float v2f;
typedef __attribute__((ext_vector_type(8))) float v8f;
typedef __attribute__((ext_vector_type(4))) unsigned int u32x4;
typedef __attribute__((ext_vector_type(8))) int i32x8;
typedef __attribute__((ext_vector_type(4))) int i32x4;

#define NBINS 256

// ---------------------------------------------------------------------------
// TDM: DMA one 16x16 f32 tile (row stride W) from global memory into LDS.
// D# group0: count=1, lds_addr, global_addr[56:0], type=2 ("image").
// D# group1: wg_mask=0 (not in cluster), data_size=2 (4B), tensor_dim0=W,
//            tensor_dim1=H, tile_dim0=tile_dim1=16, tensor_dim0_stride=W.
// Groups 2/3 zero (2D tensor). Tracked by TENSORcnt; EXEC ignored.
// 6-arg builtin form (clang-23 / therock headers): extra int32x8, zero-filled.
// ---------------------------------------------------------------------------
__device__ inline void tdm_load_tile16x16(const float* gptr, unsigned lds_off,
                                          int W, int H) {
  unsigned long long ga = (unsigned long long)(uintptr_t)gptr;
  u32x4 g0;
  g0.x = 1u;                                      // count=1, user descriptor
  g0.y = lds_off;                                 // LDS byte address
  g0.z = (unsigned)(ga & 0xFFFFFFFFull);          // global_addr[31:0]
  g0.w = (unsigned)((ga >> 32) & 0x1FFFFFFull)    // global_addr[56:32]
         | (2u << 30);                            // type = 2
  i32x8 g1;
  g1[0] = 0x00020000;                             // data_size=2 -> 4 bytes
  g1[1] = (W & 0xFFFF) << 16;                     // tensor_dim0[15:0]
  g1[2] = (W >> 16) | ((H & 0xFFFF) << 16);       // dim0[31:16] | dim1[15:0]
  g1[3] = (H >> 16) | (16 << 16);                 // dim1[31:16] | tile_dim0=16
  g1[4] = 16;                                     // tile_dim1=16, tile_dim2=0
  g1[5] = W;                                      // tensor_dim0_stride[31:0]
  g1[6] = 0;                                      // stride[47:32] | dim1_stride
  g1[7] = 0;
  i32x4 z4 = {0, 0, 0, 0};
  i32x8 z8 = {0, 0, 0, 0, 0, 0, 0, 0};
  __builtin_amdgcn_tensor_load_to_lds(g0, g1, z4, z4, z8, 0);
}

// ---------------------------------------------------------------------------
// Gram G = M * M^T via 4x V_WMMA_F32_16X16X4_F32 from an LDS-staged patch.
// TDM leaves P[k*16+m] = x[r0+k][c0+m] = M[m][k] (reference swaps axes), so
// A(m,k) = P[k*16+m], B(k,n) = P[k*16+n].
// A layout (32-bit 16x4): lanes 0-15 M=lane regs {K0,K1}; lanes 16-31 {K2,K3}.
// C/D layout: VGPR r -> row = r + 8*(lane>=16), col = lane&15.
// ---------------------------------------------------------------------------
__device__ inline void gram_wmma_lds(const float* P, int lane, float* Gout) {
  const int m  = lane & 15;
  const int hi = lane >> 4;
  v8f acc = {};
#pragma unroll
  for (int kb = 0; kb < 4; ++kb) {
    v2f av, bv;
    const int ka = 4 * kb + 2 * hi;
    av.x = P[(ka + 0) * 16 + m];
    av.y = P[(ka + 1) * 16 + m];
    bv.x = P[(4 * kb + hi) * 16 + m];
    bv.y = P[(4 * kb + 2 + hi) * 16 + m];
    acc = __builtin_amdgcn_wmma_f32_16x16x4_f32(
        /*neg_a=*/false, av, /*neg_b=*/false, bv,
        /*c_mod=*/(short)0, acc, /*reuse_a=*/false, /*reuse_b=*/false);
  }
#pragma unroll
  for (int r = 0; r < 8; ++r) Gout[(r + 8 * hi) * 16 + m] = acc[r];
}

// ---------------------------------------------------------------------------
// Cyclic Jacobi eigensolver on symmetric 16x16 G in LDS (one wave32).
// Lanes 0-15: rotate G (columns then rows); lanes 16-31: rotate V = V*J.
// ---------------------------------------------------------------------------
__device__ inline void jacobi16(float* G, float* V, int lane) {
  const int j = lane & 15;
  if (lane < 16) {
#pragma unroll
    for (int k = 0; k < 16; ++k) V[j * 16 + k] = (k == j) ? 1.0f : 0.0f;
  }
  __syncthreads();
  for (int sweep = 0; sweep < 7; ++sweep) {
    for (int p = 0; p < 15; ++p) {
      for (int q = p + 1; q < 16; ++q) {
        float app = G[p * 16 + p];
        float aqq = G[q * 16 + q];
        float apq = G[p * 16 + q];
        float c = 1.0f, s = 0.0f;
        if (fabsf(apq) > 1e-12f) {          // uniform branch (uniform reads)
          float tau = (aqq - app) / (2.0f * apq);
          float t = ((tau >= 0.0f) ? 1.0f : -1.0f) /
                    (fabsf(tau) + sqrtf(1.0f + tau * tau));
          c = rsqrtf(1.0f + t * t);
          s = t * c;
        }
        __syncthreads();
        if (lane < 16) {                    // G <- G * J   (columns p,q)
          float gjp = G[j * 16 + p], gjq = G[j * 16 + q];
          G[j * 16 + p] = c * gjp - s * gjq;
          G[j * 16 + q] = s * gjp + c * gjq;
        } else {                            // V <- V * J   (independent of G)
          float vjp = V[j * 16 + p], vjq = V[j * 16 + q];
          V[j * 16 + p] = c * vjp - s * vjq;
          V[j * 16 + q] = s * vjp + c * vjq;
        }
        __syncthreads();
        if (lane < 16) {                    // G <- J^T * G (rows p,q)
          float gpj = G[p * 16 + j], gqj = G[q * 16 + j];
          G[p * 16 + j] = c * gpj - s * gqj;
          G[q * 16 + j] = s * gpj + c * gqj;
        }
        __syncthreads();
      }
    }
  }
}

// rank-sort eigenvalues descending; emit sqrt(eig) and original-column map
__device__ inline void sort16(const float* G, float* ev, float* sOut,
                              int* cmap, int lane) {
  if (lane < 16) ev[lane] = G[lane * 16 + lane];
  __syncthreads();
  if (lane < 16) {
    float e = ev[lane];
    int r = 0;
#pragma unroll
    for (int k = 0; k < 16; ++k)
      r += (ev[k] > e) || (ev[k] == e && k < lane);
    sOut[r] = sqrtf(fmaxf(e, 0.0f));
    cmap[r] = lane;
  }
  __syncthreads();
}

// ---------------------------------------------------------------------------
// One workgroup (= one wave32) per patch-pair: TDM-stage both patches to LDS,
// Gram via WMMA, Jacobi eigensolve, singular value/vector comparison,
// binned atomic accumulate.
// ---------------------------------------------------------------------------
__global__ __launch_bounds__(32)
void svd_patch_kernel(const float* __restrict__ ref,
                      const float* __restrict__ dist,
                      int B, int H, int W, int nw, int P,
                      float invBP, float* __restrict__ bins) {
  __shared__ float Pr[256], Pd[256];
  __shared__ float G[256], Vv[256], Ur[256];
  __shared__ float ev[16], sr[16], sd[16], urd[16];
  __shared__ int   crm[16], cdm[16];

  const int lane = threadIdx.x;
  const int wg   = blockIdx.x;
  const int a    = wg / (B * P);
  const int rem  = wg - a * B * P;
  const int b    = rem / P;
  const int pidx = rem - b * P;
  const int ph   = pidx / nw;
  const int pw   = pidx - ph * nw;

  const size_t chan = (size_t)(a * B + b) * H * W;
  const size_t off  = chan + (size_t)(ph * 4) * W + (size_t)(pw * 4);

  // ---- async tensor DMA: both patches into LDS, one wait ----
  tdm_load_tile16x16(ref + off,  (unsigned)(uintptr_t)(void*)Pr, W, H);
  tdm_load_tile16x16(dist + off, (unsigned)(uintptr_t)(void*)Pd, W, H);
  __builtin_amdgcn_s_wait_tensorcnt(0);
  __syncthreads();

  // ---- reference patch: Gram (full EXEC) + eigensolve + sort ----
  gram_wmma_lds(Pr, lane, G);
  __syncthreads();
  jacobi16(G, Vv, lane);
  __syncthreads();
  sort16(G, ev, sr, crm, lane);
  for (int t = lane; t < 256; t += 32) Ur[t] = Vv[t];
  __syncthreads();

  // ---- distorted patch ----
  gram_wmma_lds(Pd, lane, G);
  __syncthreads();
  jacobi16(G, Vv, lane);
  __syncthreads();
  sort16(G, ev, sd, cdm, lane);

  // u_rd[i] = sum_j |U_r[i][j] * U_d[i][j]|  (sorted column order)
  if (lane < 16) {
    float acc = 0.0f;
#pragma unroll
    for (int k = 0; k < 16; ++k)
      acc += fabsf(Ur[lane * 16 + crm[k]] * Vv[lane * 16 + cdm[k]]);
    urd[lane] = acc;
  }
  __syncthreads();

  if (lane == 0) {
    float m = 0.0f;
    for (int k = 0; k < 16; ++k) m += urd[k];
    m *= (1.0f / 16.0f);
    float var = 0.0f;
    for (int k = 0; k < 16; ++k) { float d0 = urd[k] - m; var += d0 * d0; }
    var *= (1.0f / 15.0f);                       // ddof = 1
    float wt  = sqrtf(var) / (m + 1e-9f);
    float dss = 0.0f;
    for (int k = 0; k < 16; ++k) { float d0 = sr[k] - sd[k]; dss += d0 * d0; }
    atomicAdd(&bins[a * NBINS + (blockIdx.x & (NBINS - 1))], dss * wt * invBP);
  }
}

// ---------------------------------------------------------------------------
// Per-(a,b) channel means -> SSIM-like s1 term, accumulated per batch index.
// ---------------------------------------------------------------------------
__global__ __launch_bounds__(256)
void mean_s1_kernel(const float* __restrict__ ref,
                    const float* __restrict__ dist,
                    int B, int HW, float* __restrict__ s1sum) {
  __shared__ float br[256], bd[256];
  const int ab = blockIdx.x;
  const int a  = ab / B;
  const size_t off = (size_t)ab * HW;
  float accr = 0.0f, accd = 0.0f;
  for (int i = threadIdx.x; i < HW; i += 256) {
    accr += ref[off + i];
    accd += dist[off + i];
  }
  br[threadIdx.x] = accr;
  bd[threadIdx.x] = accd;
  __syncthreads();
  for (int st = 128; st > 0; st >>= 1) {
    if ((int)threadIdx.x < st) {
      br[threadIdx.x] += br[threadIdx.x + st];
      bd[threadIdx.x] += bd[threadIdx.x + st];
    }
    __syncthreads();
  }
  if (threadIdx.x == 0) {
    float ym = br[0] / (float)HW;   // ref mean
    float xm = bd[0] / (float)HW;   // dist mean
    float s1 = (2.0f * xm * ym + 1e-6f) / (xm * xm + ym * ym + 1e-6f);
    atomicAdd(&s1sum[a], s1);
  }
}

__global__ void zero_kernel(float* p, int n) {
  int i = blockIdx.x * blockDim.x + threadIdx.x;
  if (i < n) p[i] = 0.0f;
}

// ws layout (floats): [0,4*NBINS) layer3 bins; [4NB,8NB) layer4 bins;
// [8NB,8NB+4) s1sum layer3; [8NB+4,8NB+8) s1sum layer4
__global__ void final_kernel(const float* __restrict__ ws,
                             float* __restrict__ out) {
  int a = threadIdx.x;
  if (a >= 4) return;
  const float* b3 = ws;
  const float* b4 = ws + 4 * NBINS;
  const float* s1 = ws + 8 * NBINS;
  float ds3 = 0.0f, ds4 = 0.0f;
  for (int i = 0; i < NBINS; ++i) {
    ds3 += b3[a * NBINS + i];
    ds4 += b4[a * NBINS + i];
  }
  float g3 = expf(-2.0f * s1[a]     / 256.0f);
  float g4 = expf(-2.0f * s1[4 + a] / 512.0f);
  out[a] = ds3 * g3 + ds4 * g4;
}

extern "C" void kernel_launch(void* const* d_in, const int* in_sizes, int n_in,
                              void* d_out, int out_size, void* d_ws,
                              size_t ws_size, hipStream_t stream) {
  const float* ref3  = (const float*)d_in[0];
  const float* dist3 = (const float*)d_in[1];
  const float* ref4  = (const float*)d_in[2];
  const float* dist4 = (const float*)d_in[3];
  float* ws  = (float*)d_ws;
  float* out = (float*)d_out;

  const int wsfloats = 8 * NBINS + 8;
  zero_kernel<<<(wsfloats + 255) / 256, 256, 0, stream>>>(ws, wsfloats);

  // layer3: A=4, B=256, H=W=64, nh=nw=13, P=169
  svd_patch_kernel<<<4 * 256 * 169, 32, 0, stream>>>(
      ref3, dist3, 256, 64, 64, 13, 169, 1.0f / (256.0f * 169.0f), ws);
  // layer4: A=4, B=512, H=W=32, nh=nw=5, P=25
  svd_patch_kernel<<<4 * 512 * 25, 32, 0, stream>>>(
      ref4, dist4, 512, 32, 32, 5, 25, 1.0f / (512.0f * 25.0f),
      ws + 4 * NBINS);

  mean_s1_kernel<<<4 * 256, 256, 0, stream>>>(ref3, dist3, 256, 64 * 64,
                                              ws + 8 * NBINS);
  mean_s1_kernel<<<4 * 512, 256, 0, stream>>>(ref4, dist4, 512, 32 * 32,
                                              ws + 8 * NBINS + 4);

  final_kernel<<<1, 4, 0, stream>>>(ws, out);
}